// LocalSpatialEncoding_21148418965624
// MI455X (gfx1250) — compile-verified
//
#include <hip/hip_runtime.h>

typedef float v2f __attribute__((ext_vector_type(2)));
typedef float v8f __attribute__((ext_vector_type(8)));
typedef float f4  __attribute__((ext_vector_type(4)));

#define BN_EPS 1e-5f
constexpr int Bb   = 2;
constexpr int Npts = 65536;
constexpr int Kn   = 16;
constexpr int Dch  = 32;

// Fold conv weight + bias + BN into an effective 32x8 weight Wf acting on
// [cx,cy,cz, nx,ny,nz, dist, 1]:
//   y = relu( Wf @ v ),  Wf = s*[Wc+Wd, Wn-Wd, w_dist],  col7 = s*(b-mean)+beta
__global__ void prep_weights(const float* __restrict__ w, const float* __restrict__ bias,
                             const float* __restrict__ gamma, const float* __restrict__ beta,
                             const float* __restrict__ mean, const float* __restrict__ var,
                             float* __restrict__ wf) {
    int c = threadIdx.x;
    if (c >= Dch) return;
    float s = gamma[c] * rsqrtf(var[c] + BN_EPS);
    const float* wr = w + c * 10;     // [center(3), neighbor(3), diff(3), dist(1)]
    wf[c*8 + 0] = s * (wr[0] + wr[6]);
    wf[c*8 + 1] = s * (wr[1] + wr[7]);
    wf[c*8 + 2] = s * (wr[2] + wr[8]);
    wf[c*8 + 3] = s * (wr[3] - wr[6]);
    wf[c*8 + 4] = s * (wr[4] - wr[7]);
    wf[c*8 + 5] = s * (wr[5] - wr[8]);
    wf[c*8 + 6] = s * wr[9];
    wf[c*8 + 7] = s * (bias[c] - mean[c]) + beta[c];
}

__global__ __launch_bounds__(256) void lse_wmma_kernel(
        const float* __restrict__ coords,   // (B, N, 3)
        const float* __restrict__ feats,    // (B, 32, N, 1)
        const int*   __restrict__ idx,      // (B, N, 16)
        const float* __restrict__ wf,       // (32, 8) folded weights
        float*       __restrict__ out) {    // (B, 64, N, 16)
    const int lane = threadIdx.x & 31;
    const int wid  = threadIdx.x >> 5;
    const int m    = lane & 15;             // A-row (= neighbor k) / B,D column (= channel)
    const int half = lane >> 4;             // selects K-col pair / D row-half

    const int pt = blockIdx.x * 8 + wid;    // one point per wave (uniform per wave)
    if (pt >= Bb * Npts) return;
    const int b = pt >> 16;                 // Npts == 2^16
    const int n = pt & (Npts - 1);

    // ---- gather geometry ----
    const float* cp = coords + ((size_t)b * Npts + n) * 3;   // broadcast load
    float cx = cp[0], cy = cp[1], cz = cp[2];

    int j = idx[((size_t)b * Npts + n) * Kn + m];            // neighbor index for k = m
    const float* qp = coords + ((size_t)b * Npts + j) * 3;   // L2-resident gather (1.5 MB)
    float nx = qp[0], ny = qp[1], nz = qp[2];

    float dx = cx - nx, dy = cy - ny, dz = cz - nz;
    float dist = sqrtf(dx*dx + dy*dy + dz*dz);

    // ---- A fragments: 16x4 f32 layout => lane 16h+m holds row m, cols 2h..2h+1
    // step0 cols 0-3 = [cx,cy,cz,nx] ; step1 cols 4-7 = [ny,nz,dist,1]
    v2f A0, A1;
    A0.x = half ? cz   : cx;   A0.y = half ? nx  : cy;
    A1.x = half ? dist : ny;   A1.y = half ? 1.f : nz;

    // ---- B fragments: 4x16 f32, lane holds channel c=m (+16 for tile1), K-cols 4s+2h+{0,1}
    const float* w0 = wf + (size_t)m        * 8 + 2 * half;
    const float* w1 = wf + (size_t)(m + 16) * 8 + 2 * half;
    v2f B00 = { w0[0], w0[1] };
    v2f B01 = { w0[4], w0[5] };
    v2f B10 = { w1[0], w1[1] };
    v2f B11 = { w1[4], w1[5] };

    // ---- D = A x B (16 k-rows x 16 channels), two channel tiles, K=8 in two steps
    v8f acc0 = {}, acc1 = {};
    acc0 = __builtin_amdgcn_wmma_f32_16x16x4_f32(false, A0, false, B00, (short)0, acc0, false, false);
    acc0 = __builtin_amdgcn_wmma_f32_16x16x4_f32(false, A1, false, B01, (short)0, acc0, false, false);
    acc1 = __builtin_amdgcn_wmma_f32_16x16x4_f32(false, A0, false, B10, (short)0, acc1, false, false);
    acc1 = __builtin_amdgcn_wmma_f32_16x16x4_f32(false, A1, false, B11, (short)0, acc1, false, false);

#pragma unroll
    for (int r = 0; r < 8; ++r) {
        acc0[r] = fmaxf(acc0[r], 0.f);
        acc1[r] = fmaxf(acc1[r], 0.f);
    }

    // ---- store y: lane holds channel c=m(+16), k = 8*half + 0..7 -> 32B contiguous runs
    {
        size_t base0 = (((size_t)b * 64 + m     ) * Npts + n) * Kn + 8 * half;
        size_t base1 = (((size_t)b * 64 + 16 + m) * Npts + n) * Kn + 8 * half;
        f4* p0 = (f4*)(out + base0);
        f4* p1 = (f4*)(out + base1);
        f4 lo0 = { acc0[0], acc0[1], acc0[2], acc0[3] };
        f4 hi0 = { acc0[4], acc0[5], acc0[6], acc0[7] };
        f4 lo1 = { acc1[0], acc1[1], acc1[2], acc1[3] };
        f4 hi1 = { acc1[4], acc1[5], acc1[6], acc1[7] };
        __builtin_nontemporal_store(lo0, p0);
        __builtin_nontemporal_store(hi0, p0 + 1);
        __builtin_nontemporal_store(lo1, p1);
        __builtin_nontemporal_store(hi1, p1 + 1);
    }

    // ---- feature broadcast: lane = channel, splat over K=16
    {
        float f = feats[((size_t)b * Dch + lane) * Npts + n];
        f4 v = { f, f, f, f };
        f4* p = (f4*)(out + (((size_t)b * 64 + 32 + lane) * Npts + n) * Kn);
        __builtin_nontemporal_store(v, p);
        __builtin_nontemporal_store(v, p + 1);
        __builtin_nontemporal_store(v, p + 2);
        __builtin_nontemporal_store(v, p + 3);
    }
}

extern "C" void kernel_launch(void* const* d_in, const int* in_sizes, int n_in,
                              void* d_out, int out_size, void* d_ws, size_t ws_size,
                              hipStream_t stream) {
    const float* coords   = (const float*)d_in[0];
    const float* features = (const float*)d_in[1];
    const int*   idx      = (const int*)  d_in[2];
    const float* w        = (const float*)d_in[3];
    const float* b        = (const float*)d_in[4];
    const float* gamma    = (const float*)d_in[5];
    const float* beta     = (const float*)d_in[6];
    const float* mean     = (const float*)d_in[7];
    const float* var      = (const float*)d_in[8];

    float* wf = (float*)d_ws;   // 32*8 floats

    prep_weights<<<1, 32, 0, stream>>>(w, b, gamma, beta, mean, var, wf);

    const int total_waves = Bb * Npts;          // one wave per point
    const int blocks = total_waves / 8;         // 256 threads = 8 waves per block
    lse_wmma_kernel<<<blocks, 256, 0, stream>>>(coords, features, idx, wf, (float*)d_out);
}